// GNNEncoder_90142773608475
// MI455X (gfx1250) — compile-verified
//
#include <hip/hip_runtime.h>
#include <math.h>

// Problem constants (match reference)
#define NVN 100000   // multiple of 16 -> no partial M tiles in GEMM
#define NCN 50000    // multiple of 16
#define EN  400000
// HID=128, HEADS=4, CH=32, B=16, LV=22, LC=64

typedef float v2f __attribute__((ext_vector_type(2)));
typedef float v8f __attribute__((ext_vector_type(8)));

// ---------------------------------------------------------------------------
// utility kernels
// ---------------------------------------------------------------------------
__global__ __launch_bounds__(256) void fill_kernel(float* __restrict__ p, int n, float v) {
    int i = blockIdx.x * 256 + threadIdx.x;
    if (i < n) p[i] = v;
}

__device__ __forceinline__ void atomicMaxF(float* addr, float val) {
    unsigned int* ua = (unsigned int*)addr;
    unsigned int cur = *ua;
    while (__uint_as_float(cur) < val) {
        unsigned int prev = atomicCAS(ua, cur, __float_as_uint(val));
        if (prev == cur) break;
        cur = prev;
    }
}

// ---------------------------------------------------------------------------
// Fourier feature lift
// xv[n, s*3+f]      = sin(x_f / 2^(s-11)),  s in [0,22), f in {obj,lb,ub}
// xv[n, 66+s*3+f]   = cos(...)
// ---------------------------------------------------------------------------
__global__ __launch_bounds__(256) void fourier_v_kernel(
    const float* __restrict__ obj, const float* __restrict__ lb,
    const float* __restrict__ ub, float* __restrict__ xv) {
    int idx = blockIdx.x * 256 + threadIdx.x;
    if (idx >= NVN * 66) return;
    int n = idx / 66, j = idx - n * 66;
    int s = j / 3, f = j - s * 3;
    float x;
    if (f == 0) {
        x = obj[n];
    } else {
        float u = ub[n];
        if (f == 2) {                              // ub: only +inf fixed
            x = (isinf(u) && u > 0.f) ? 100.f : u;
        } else {                                   // lb: +inf->100, -inf->-100, and ub==-inf masks lb
            float l2 = lb[n];
            if (isinf(l2)) l2 = (l2 > 0.f) ? 100.f : -100.f;
            if (isinf(u) && u < 0.f) l2 = -100.f;
            x = l2;
        }
    }
    float y = x * exp2f(11.0f - (float)s);         // x / 2^(s-11)
    xv[n * 132 + j]      = sinf(y);
    xv[n * 132 + 66 + j] = cosf(y);
}

__global__ __launch_bounds__(256) void fourier_c_kernel(
    const float* __restrict__ clb, float* __restrict__ xc) {
    int idx = blockIdx.x * 256 + threadIdx.x;
    if (idx >= NCN * 64) return;
    int n = idx >> 6, j = idx & 63;
    float y = clb[n] * exp2f(32.0f - (float)j);    // clb / 2^(j-32)
    xc[n * 128 + j]      = sinf(y);
    xc[n * 128 + 64 + j] = cosf(y);
}

// ---------------------------------------------------------------------------
// Y[M,128] = X[M,K] @ W[K,128] + bias   via V_WMMA_F32_16X16X4_F32
// K is a compile-time constant (132 or 128) -> fully unrolled load/WMMA stream
// with immediate offsets, no per-iteration SALU, no EXEC-mask churn.
// M is guaranteed a multiple of 16 (NVN, NCN) -> no row guards anywhere.
// One wave32 per 16x16 output tile; block = 8 waves = one 16x128 row band.
// A frag: lane(l) row m=l&15, khalf=2*(l>>4); vgpr j -> A[m][khalf+j]  (float2)
// B frag: lane(l) col n=l&15; vgpr j -> W[khalf+j][n]
// D frag: vgpr r -> (m = r + 8*(l>>4), n = l&15)
// ---------------------------------------------------------------------------
template <int K>
__global__ __launch_bounds__(256) void gemm_bias_wmma(
    const float* __restrict__ X, const float* __restrict__ W,
    const float* __restrict__ bias, float* __restrict__ Y) {
    const int lane  = threadIdx.x & 31;
    const int tn    = threadIdx.x >> 5;            // 0..7  (N tile, N=128)
    const int tm    = blockIdx.x;                  // M tile
    const int mrow  = tm * 16 + (lane & 15);
    const int khalf = (lane >> 4) << 1;            // 0 or 2
    const int ncol  = tn * 16 + (lane & 15);
    const float* xrow = X + (size_t)mrow * K + khalf;
    const float* wcol = W + (size_t)khalf * 128 + ncol;

    v8f c = {};
#pragma unroll
    for (int kk = 0; kk < K; kk += 4) {
        float2 t = *(const float2*)(xrow + kk);
        v2f a; a[0] = t.x; a[1] = t.y;
        v2f b;
        b[0] = wcol[(size_t)kk * 128];
        b[1] = wcol[(size_t)kk * 128 + 128];
        c = __builtin_amdgcn_wmma_f32_16x16x4_f32(
                false, a, false, b, (short)0, c, false, false);
    }
    const int msub = (lane >> 4) << 3;             // 0 or 8
    const float bv = bias[ncol];
    float* yp = Y + (size_t)(tm * 16 + msub) * 128 + ncol;
#pragma unroll
    for (int r = 0; r < 8; ++r) {
        yp[(size_t)r * 128] = c[r] + bv;
    }
}

// ---------------------------------------------------------------------------
// acc[n,j] = b1[j] (+ b2[j])   (seed scatter accumulator with GAT biases)
// ---------------------------------------------------------------------------
__global__ __launch_bounds__(256) void init_acc_kernel(
    float* __restrict__ acc, const float* __restrict__ b1,
    const float* __restrict__ b2, int N) {
    int idx = blockIdx.x * 256 + threadIdx.x;
    if (idx >= N * 128) return;
    int j = idx & 127;
    float v = b1[j];
    if (b2) v += b2[j];
    acc[idx] = v;
}

// ---------------------------------------------------------------------------
// Edge pass A: alpha[e,h] = sum_c leaky(xl[src]+xr[dst]+ev*We, 0.2)*att[h,c]
// one wave per edge; lane t covers channels 4t..4t+3 (head = t>>3);
// 8-lane shuffle reduction per head; segment-max via CAS atomic.
// ---------------------------------------------------------------------------
__global__ __launch_bounds__(256) void edge_alpha_kernel(
    const int* __restrict__ src, const int* __restrict__ dst,
    const float* __restrict__ ev,
    const float* __restrict__ xl, const float* __restrict__ xr,
    const float* __restrict__ We, const float* __restrict__ att,
    float* __restrict__ alpha, float* __restrict__ amax) {
    int e = blockIdx.x * 8 + (threadIdx.x >> 5);
    if (e >= EN) return;
    int t = threadIdx.x & 31;
    int s = src[e], d = dst[e];
    float evv = ev[e];
    float4 a4 = *(const float4*)(xl + (size_t)s * 128 + t * 4);
    float4 b4 = *(const float4*)(xr + (size_t)d * 128 + t * 4);
    float4 w4 = *(const float4*)(We + t * 4);
    float4 t4 = *(const float4*)(att + t * 4);
    float m0 = a4.x + b4.x + evv * w4.x; m0 = (m0 > 0.f) ? m0 : 0.2f * m0;
    float m1 = a4.y + b4.y + evv * w4.y; m1 = (m1 > 0.f) ? m1 : 0.2f * m1;
    float m2 = a4.z + b4.z + evv * w4.z; m2 = (m2 > 0.f) ? m2 : 0.2f * m2;
    float m3 = a4.w + b4.w + evv * w4.w; m3 = (m3 > 0.f) ? m3 : 0.2f * m3;
    float p = m0 * t4.x + m1 * t4.y + m2 * t4.z + m3 * t4.w;
    p += __shfl_xor(p, 4, 8);
    p += __shfl_xor(p, 2, 8);
    p += __shfl_xor(p, 1, 8);
    if ((t & 7) == 0) {
        int h = t >> 3;
        alpha[e * 4 + h] = p;
        atomicMaxF(&amax[(size_t)d * 4 + h], p);
    }
}

// Edge pass B: ex = exp(alpha - amax[dst]);  denom[dst,h] += ex
__global__ __launch_bounds__(256) void edge_exp_kernel(
    const int* __restrict__ dst, const float* __restrict__ alpha,
    const float* __restrict__ amax, float* __restrict__ exb,
    float* __restrict__ denom) {
    int idx = blockIdx.x * 256 + threadIdx.x;
    if (idx >= EN * 4) return;
    int e = idx >> 2, h = idx & 3;
    int d = dst[e];
    float ex = __expf(alpha[idx] - amax[(size_t)d * 4 + h]);
    exb[idx] = ex;
    atomicAdd(&denom[(size_t)d * 4 + h], ex);
}

// Edge pass C: acc[dst, h*32+c] += (ex/(denom+1e-16)) * xl[src, h*32+c]
__global__ __launch_bounds__(256) void edge_scatter_kernel(
    const int* __restrict__ src, const int* __restrict__ dst,
    const float* __restrict__ exb, const float* __restrict__ denom,
    const float* __restrict__ xl, float* __restrict__ acc) {
    int e = blockIdx.x * 8 + (threadIdx.x >> 5);
    if (e >= EN) return;
    int t = threadIdx.x & 31;
    int h = t >> 3;
    int s = src[e], d = dst[e];
    float a = exb[e * 4 + h] / (denom[(size_t)d * 4 + h] + 1e-16f);
    float4 x4 = *(const float4*)(xl + (size_t)s * 128 + t * 4);
    float* ap = acc + (size_t)d * 128 + t * 4;
    atomicAdd(ap + 0, a * x4.x);
    atomicAdd(ap + 1, a * x4.y);
    atomicAdd(ap + 2, a * x4.z);
    atomicAdd(ap + 3, a * x4.w);
}

// ---------------------------------------------------------------------------
// GraphNorm (B=16 segments): three passes over [N,128]
// ---------------------------------------------------------------------------
__global__ __launch_bounds__(256) void gn_pass1(
    const float* __restrict__ x, const int* __restrict__ batch,
    float* __restrict__ sum, float* __restrict__ cnt, int N) {
    int idx = blockIdx.x * 256 + threadIdx.x;
    if (idx >= N * 128) return;
    int n = idx >> 7, j = idx & 127;
    int b = batch[n];
    atomicAdd(&sum[b * 128 + j], x[idx]);
    if (j == 0) atomicAdd(&cnt[b], 1.0f);
}

__global__ __launch_bounds__(256) void gn_pass2(
    float* __restrict__ x, const int* __restrict__ batch,
    const float* __restrict__ sum, const float* __restrict__ cnt,
    const float* __restrict__ ms, float* __restrict__ var, int N) {
    int idx = blockIdx.x * 256 + threadIdx.x;
    if (idx >= N * 128) return;
    int n = idx >> 7, j = idx & 127;
    int b = batch[n];
    float c = fmaxf(cnt[b], 1.f);
    float mean = sum[b * 128 + j] / c;
    float o = x[idx] - mean * ms[j];
    x[idx] = o;
    atomicAdd(&var[b * 128 + j], o * o);
}

__global__ __launch_bounds__(256) void gn_pass3(
    const float* __restrict__ x, const int* __restrict__ batch,
    const float* __restrict__ var, const float* __restrict__ cnt,
    const float* __restrict__ w, const float* __restrict__ bb,
    float* __restrict__ y, int N) {
    int idx = blockIdx.x * 256 + threadIdx.x;
    if (idx >= N * 128) return;
    int n = idx >> 7, j = idx & 127;
    int b = batch[n];
    float c = fmaxf(cnt[b], 1.f);
    float stdv = sqrtf(var[b * 128 + j] / c + 1e-5f);
    float r = w[j] * x[idx] / stdv + bb[j];
    y[idx] = (r > 0.f) ? r : 0.01f * r;       // leaky_relu 0.01
}

// ---------------------------------------------------------------------------
// host driver
// ---------------------------------------------------------------------------
extern "C" void kernel_launch(void* const* d_in, const int* in_sizes, int n_in,
                              void* d_out, int out_size, void* d_ws, size_t ws_size,
                              hipStream_t stream) {
    (void)in_sizes; (void)n_in; (void)out_size; (void)ws_size;
    const float* obj = (const float*)d_in[0];
    const float* lb  = (const float*)d_in[1];
    const float* ub  = (const float*)d_in[2];
    const float* clb = (const float*)d_in[3];
    const float* ev  = (const float*)d_in[4];

    // params flattened in jax pytree (sorted-key) order:
    // gn{b,ms,w}, then layers[0..2] x {cv,vc,vv} x {We,Wl,Wr,att,bias,bl,br}
    int pi = 5;
    const float* gn_b  = (const float*)d_in[pi++];
    const float* gn_ms = (const float*)d_in[pi++];
    const float* gn_w  = (const float*)d_in[pi++];
    struct Rel { const float *We, *Wl, *Wr, *att, *bias, *bl, *br; };
    Rel R[3][3];  // [layer][0=cv, 1=vc, 2=vv]
    for (int l = 0; l < 3; ++l)
        for (int r = 0; r < 3; ++r) {
            R[l][r].We   = (const float*)d_in[pi++];
            R[l][r].Wl   = (const float*)d_in[pi++];
            R[l][r].Wr   = (const float*)d_in[pi++];
            R[l][r].att  = (const float*)d_in[pi++];
            R[l][r].bias = (const float*)d_in[pi++];
            R[l][r].bl   = (const float*)d_in[pi++];
            R[l][r].br   = (const float*)d_in[pi++];
        }
    const int* rows = (const int*)d_in[pi++];  // in [0,NC)
    const int* cols = (const int*)d_in[pi++];  // in [0,NV)
    const int* vb   = (const int*)d_in[pi++];
    const int* cb   = (const int*)d_in[pi++];

    // workspace layout (floats)
    float* ws = (float*)d_ws;
    size_t o = 0;
    float* xv    = ws + o; o += (size_t)NVN * 132;
    float* xc    = ws + o; o += (size_t)NCN * 128;
    float* xlb   = ws + o; o += (size_t)NVN * 128;
    float* xrb   = ws + o; o += (size_t)NVN * 128;
    float* accV  = ws + o; o += (size_t)NVN * 128;
    float* accC  = ws + o; o += (size_t)NCN * 128;
    float* alpha = ws + o; o += (size_t)EN * 4;
    float* exb   = ws + o; o += (size_t)EN * 4;
    float* amax  = ws + o; o += (size_t)NVN * 4;
    float* denom = ws + o; o += (size_t)NVN * 4;
    float* gsum  = ws + o; o += 16 * 128;
    float* gvar  = ws + o; o += 16 * 128;
    float* gcnt  = ws + o; o += 16;

    float* out = (float*)d_out;
    const dim3 T(256);

    auto fill = [&](float* p, int n, float v) {
        fill_kernel<<<dim3((n + 255) / 256), T, 0, stream>>>(p, n, v);
    };
    auto gemm = [&](const float* X, const float* W, const float* bias,
                    float* Y, int M, int K) {
        if (K == 132)
            gemm_bias_wmma<132><<<dim3(M / 16), T, 0, stream>>>(X, W, bias, Y);
        else
            gemm_bias_wmma<128><<<dim3(M / 16), T, 0, stream>>>(X, W, bias, Y);
    };
    auto edges = [&](const int* s, const int* d, const Rel& r,
                     const float* xlp, const float* xrp, int ndst, float* acc) {
        fill(amax, ndst * 4, -1.0e30f);
        fill(denom, ndst * 4, 0.f);
        edge_alpha_kernel<<<dim3(EN / 8), T, 0, stream>>>(s, d, ev, xlp, xrp,
                                                          r.We, r.att, alpha, amax);
        edge_exp_kernel<<<dim3((EN * 4) / 256), T, 0, stream>>>(d, alpha, amax, exb, denom);
        edge_scatter_kernel<<<dim3(EN / 8), T, 0, stream>>>(s, d, exb, denom, xlp, acc);
    };
    auto gnorm = [&](float* x, const int* batch, int N, float* y) {
        fill(gsum, 16 * 128 * 2 + 16, 0.f);  // gsum, gvar, gcnt are contiguous
        int blks = (N * 128 + 255) / 256;
        gn_pass1<<<dim3(blks), T, 0, stream>>>(x, batch, gsum, gcnt, N);
        gn_pass2<<<dim3(blks), T, 0, stream>>>(x, batch, gsum, gcnt, gn_ms, gvar, N);
        gn_pass3<<<dim3(blks), T, 0, stream>>>(x, batch, gvar, gcnt, gn_w, gn_b, y, N);
    };

    // Fourier lift
    fourier_v_kernel<<<dim3((NVN * 66 + 255) / 256), T, 0, stream>>>(obj, lb, ub, xv);
    fourier_c_kernel<<<dim3((NCN * 64 + 255) / 256), T, 0, stream>>>(clb, xc);

    for (int l = 0; l < 3; ++l) {
        const int Kv = (l == 0) ? 132 : 128;
        const Rel& cv = R[l][0];
        const Rel& vc = R[l][1];
        const Rel& vv = R[l][2];

        // ---- nv = gatv2_vv(xv->xv over (rows->cols)) + gatv2_cv(xc->xv) ----
        init_acc_kernel<<<dim3((NVN * 128 + 255) / 256), T, 0, stream>>>(
            accV, vv.bias, cv.bias, NVN);
        // vv relation: src indexes xl (NV rows), dst = cols, num_dst = NV
        gemm(xv, vv.Wl, vv.bl, xlb, NVN, Kv);
        gemm(xv, vv.Wr, vv.br, xrb, NVN, Kv);
        edges(rows, cols, vv, xlb, xrb, NVN, accV);
        // cv relation: xl from xc (NC rows), xr from xv
        gemm(xc, cv.Wl, cv.bl, xlb, NCN, 128);
        gemm(xv, cv.Wr, cv.br, xrb, NVN, Kv);
        edges(rows, cols, cv, xlb, xrb, NVN, accV);

        // ---- nc = gatv2_vc(xv->xc over (cols->rows)) ----
        init_acc_kernel<<<dim3((NCN * 128 + 255) / 256), T, 0, stream>>>(
            accC, vc.bias, nullptr, NCN);
        gemm(xv, vc.Wl, vc.bl, xlb, NVN, Kv);
        gemm(xc, vc.Wr, vc.br, xrb, NCN, 128);
        edges(cols, rows, vc, xlb, xrb, NCN, accC);

        // ---- GraphNorm + leaky_relu(0.01); last layer writes d_out ----
        float* xvDst = (l == 2) ? out : xv;                       // xv becomes [NV,128]
        float* xcDst = (l == 2) ? out + (size_t)NVN * 128 : xc;
        gnorm(accV, vb, NVN, xvDst);
        gnorm(accC, cb, NCN, xcDst);
    }
}